// RGINConv_51762945852038
// MI455X (gfx1250) — compile-verified
//
#include <hip/hip_runtime.h>
#include <hip/hip_bf16.h>

// R-GIN on MI455X (gfx1250). FP32 WMMA 16x16x4 for all three GEMM stages,
// L2-resident f32 atomics for the edge scatter (102.4MB agg << 192MB L2),
// fused BN statistics in the GEMM1 epilogue, in-place agg->h1 reuse.
// Epilogues use a block-uniform full-strip fast path (781/782 blocks) so the
// common case is a branch-free global_store_b32 stream.

constexpr int NN = 50000;    // nodes
constexpr int NE = 600000;   // edges
constexpr int FD = 128;      // features (in == out)
constexpr int NR = 4;        // relations
constexpr float BN_EPS = 1e-5f;

constexpr int ROWS = 64;     // node rows per workgroup strip
constexpr int LDA  = 132;    // padded LDS row stride (floats) -> spreads 64 banks
constexpr int LDS_FLOATS = ROWS * LDA + FD * LDA;   // A strip + W tile (~101 KB)

typedef __attribute__((ext_vector_type(2))) float v2f;
typedef __attribute__((ext_vector_type(8))) float v8f;

// ---------------------------------------------------------------- zero ----
__global__ __launch_bounds__(256) void zero_kernel(float4* __restrict__ p, long n4) {
  long i = (long)blockIdx.x * blockDim.x + threadIdx.x;
  if (i < n4) p[i] = make_float4(0.f, 0.f, 0.f, 0.f);
}

// ------------------------------------------------------------- scatter ----
// One edge per wave32: lane holds 4 features (float4 gather of x[src]),
// 4 global f32 atomics into agg[rel, dst, :]. All RMW stays in L2.
__global__ __launch_bounds__(256) void scatter_kernel(const float* __restrict__ x,
                                                      const int* __restrict__ ei,
                                                      const int* __restrict__ et,
                                                      float* __restrict__ buf) {
  const int wave = threadIdx.x >> 5;
  const int lane = threadIdx.x & 31;
  const long e = (long)blockIdx.x * 8 + wave;
  if (e >= NE) return;
  const int dst = ei[e];
  const int src = ei[NE + e];
  const int r   = et[e];
  const float4 v = ((const float4*)(x + (size_t)src * FD))[lane];
  float* d = buf + ((size_t)r * NN + dst) * FD + lane * 4;
  atomicAdd(d + 0, v.x);
  atomicAdd(d + 1, v.y);
  atomicAdd(d + 2, v.z);
  atomicAdd(d + 3, v.w);
}

// --------------------------------------------------------------- gemm1 ----
// h1[r] = (x + agg[r]) @ W1[r] + b1[r]; stored in place over agg.
// Fused BN partial sums (sum, sumsq) via cross-half shuffle + 1 atomic/col.
__global__ __launch_bounds__(256) void gemm1_kernel(const float* __restrict__ x,
                                                    const float* __restrict__ W1,
                                                    const float* __restrict__ b1,
                                                    float* __restrict__ buf,
                                                    float* __restrict__ sums,
                                                    float* __restrict__ sumsq) {
  extern __shared__ float smem[];
  float* As = smem;                 // ROWS x LDA
  float* Ws = smem + ROWS * LDA;    // FD x LDA
  const int r    = blockIdx.y;
  const int base = blockIdx.x * ROWS;
  const int tid  = threadIdx.x;

  // Stage W1[r] (128x128) into LDS with padded stride.
  const float4* wsrc = (const float4*)(W1 + (size_t)r * FD * FD);
  for (int i = tid; i < FD * FD / 4; i += 256) {
    const int k = i >> 5, n4 = i & 31;
    const float4 w = wsrc[(size_t)k * 32 + n4];
    float* d = &Ws[k * LDA + n4 * 4];
    d[0] = w.x; d[1] = w.y; d[2] = w.z; d[3] = w.w;
  }
  // Stage A = x + agg for this 64-row strip (row-clamped for the tail WG).
  for (int i = tid; i < ROWS * FD / 4; i += 256) {
    const int rl = i >> 5, k4 = i & 31;
    int row = base + rl; if (row >= NN) row = NN - 1;
    const float4 a = ((const float4*)(x   + (size_t)row * FD))[k4];
    const float4 g = ((const float4*)(buf + ((size_t)r * NN + row) * FD))[k4];
    float* d = &As[rl * LDA + k4 * 4];
    d[0] = a.x + g.x; d[1] = a.y + g.y; d[2] = a.z + g.z; d[3] = a.w + g.w;
  }
  __syncthreads();

  const int wave = tid >> 5, lane = tid & 31;
  const int half = lane >> 4, l = lane & 15;
  const int n = wave * 16 + l;          // wave owns a 16-wide column tile
  v8f acc[4]; const v8f zz = {};
  acc[0] = zz; acc[1] = zz; acc[2] = zz; acc[3] = zz;

  for (int k0 = 0; k0 < FD; k0 += 4) {
    const int kk = k0 + 2 * half;       // B frag: VGPR0 row kk, VGPR1 row kk+1
    v2f b;
    b.x = Ws[kk * LDA + n];
    b.y = Ws[(kk + 1) * LDA + n];
#pragma unroll
    for (int rt = 0; rt < 4; ++rt) {    // A frag: row = lane&15 (+16*rt), K = kk,kk+1
      const v2f a = *(const v2f*)&As[(rt * 16 + l) * LDA + kk];
      acc[rt] = __builtin_amdgcn_wmma_f32_16x16x4_f32(
          false, a, false, b, (short)0, acc[rt], false, false);
    }
  }

  const float bias = b1[r * FD + n];
  float s = 0.f, s2 = 0.f;
  float* hout = buf + ((size_t)r * NN + base + 8 * half) * FD + n;

  if (base + ROWS <= NN) {
    // Fast path: whole 64-row strip in range -> branch-free stores.
#pragma unroll
    for (int rt = 0; rt < 4; ++rt) {
#pragma unroll
      for (int v = 0; v < 8; ++v) {
        const float val = acc[rt][v] + bias;
        hout[(size_t)(rt * 16 + v) * FD] = val;
        s += val; s2 += val * val;
      }
    }
  } else {
#pragma unroll
    for (int rt = 0; rt < 4; ++rt) {
#pragma unroll
      for (int v = 0; v < 8; ++v) {
        const int row = base + rt * 16 + v + 8 * half;   // C layout: M = vgpr (+8 hi)
        const float val = acc[rt][v] + bias;
        if (row < NN) {
          buf[((size_t)r * NN + row) * FD + n] = val;
          s += val; s2 += val * val;
        }
      }
    }
  }
  s  += __shfl_xor(s, 16, 32);
  s2 += __shfl_xor(s2, 16, 32);
  if (half == 0) {
    atomicAdd(&sums[r * FD + n], s);
    atomicAdd(&sumsq[r * FD + n], s2);
  }
}

// --------------------------------------------------------- bn finalize ----
__global__ __launch_bounds__(256) void bn_finalize_kernel(const float* __restrict__ sums,
                                                          const float* __restrict__ sumsq,
                                                          const float* __restrict__ gamma,
                                                          const float* __restrict__ beta,
                                                          float* __restrict__ scale,
                                                          float* __restrict__ shift) {
  const int i = blockIdx.x * blockDim.x + threadIdx.x;
  if (i < NR * FD) {
    const float inv_n = 1.0f / (float)NN;
    const float mean = sums[i] * inv_n;
    const float var  = sumsq[i] * inv_n - mean * mean;
    const float sc   = gamma[i] * rsqrtf(var + BN_EPS);
    scale[i] = sc;
    shift[i] = beta[i] - mean * sc;
  }
}

// --------------------------------------------------------------- gemm2 ----
// out = x@W_self + b_self + sum_r ( relu(bn(h1[r])) @ W2[r] + b2[r] )
// Accumulators live in VGPRs across 5 phases (W_self, then r=0..3).
__global__ __launch_bounds__(256) void gemm2_kernel(const float* __restrict__ x,
                                                    const float* __restrict__ W_self,
                                                    const float* __restrict__ b_self,
                                                    const float* __restrict__ W2,
                                                    const float* __restrict__ b2,
                                                    const float* __restrict__ buf,
                                                    const float* __restrict__ scale,
                                                    const float* __restrict__ shift,
                                                    float* __restrict__ out) {
  extern __shared__ float smem[];
  float* As = smem;
  float* Ws = smem + ROWS * LDA;
  const int base = blockIdx.x * ROWS;
  const int tid  = threadIdx.x;
  const int wave = tid >> 5, lane = tid & 31;
  const int half = lane >> 4, l = lane & 15;
  const int n = wave * 16 + l;

  v8f acc[4]; const v8f zz = {};
  acc[0] = zz; acc[1] = zz; acc[2] = zz; acc[3] = zz;

  for (int phase = 0; phase < 5; ++phase) {
    const float* wsrcf = (phase == 0) ? W_self : (W2 + (size_t)(phase - 1) * FD * FD);
    const float4* wsrc = (const float4*)wsrcf;
    for (int i = tid; i < FD * FD / 4; i += 256) {
      const int k = i >> 5, n4 = i & 31;
      const float4 w = wsrc[(size_t)k * 32 + n4];
      float* d = &Ws[k * LDA + n4 * 4];
      d[0] = w.x; d[1] = w.y; d[2] = w.z; d[3] = w.w;
    }
    if (phase == 0) {
      for (int i = tid; i < ROWS * FD / 4; i += 256) {
        const int rl = i >> 5, k4 = i & 31;
        int row = base + rl; if (row >= NN) row = NN - 1;
        const float4 a = ((const float4*)(x + (size_t)row * FD))[k4];
        float* d = &As[rl * LDA + k4 * 4];
        d[0] = a.x; d[1] = a.y; d[2] = a.z; d[3] = a.w;
      }
    } else {
      const int r = phase - 1;
      for (int i = tid; i < ROWS * FD / 4; i += 256) {
        const int rl = i >> 5, k4 = i & 31;
        int row = base + rl; if (row >= NN) row = NN - 1;
        const float4 h = ((const float4*)(buf + ((size_t)r * NN + row) * FD))[k4];
        const int k = k4 * 4;
        float* d = &As[rl * LDA + k];
        d[0] = fmaxf(h.x * scale[r * FD + k + 0] + shift[r * FD + k + 0], 0.f);
        d[1] = fmaxf(h.y * scale[r * FD + k + 1] + shift[r * FD + k + 1], 0.f);
        d[2] = fmaxf(h.z * scale[r * FD + k + 2] + shift[r * FD + k + 2], 0.f);
        d[3] = fmaxf(h.w * scale[r * FD + k + 3] + shift[r * FD + k + 3], 0.f);
      }
    }
    __syncthreads();

    for (int k0 = 0; k0 < FD; k0 += 4) {
      const int kk = k0 + 2 * half;
      v2f b;
      b.x = Ws[kk * LDA + n];
      b.y = Ws[(kk + 1) * LDA + n];
#pragma unroll
      for (int rt = 0; rt < 4; ++rt) {
        const v2f a = *(const v2f*)&As[(rt * 16 + l) * LDA + kk];
        acc[rt] = __builtin_amdgcn_wmma_f32_16x16x4_f32(
            false, a, false, b, (short)0, acc[rt], false, false);
      }
    }
    __syncthreads();   // LDS reused next phase
  }

  const float bias = b_self[n] + b2[0 * FD + n] + b2[1 * FD + n]
                   + b2[2 * FD + n] + b2[3 * FD + n];
  float* oout = out + (size_t)(base + 8 * half) * FD + n;

  if (base + ROWS <= NN) {
    // Fast path: branch-free output stream.
#pragma unroll
    for (int rt = 0; rt < 4; ++rt) {
#pragma unroll
      for (int v = 0; v < 8; ++v) {
        oout[(size_t)(rt * 16 + v) * FD] = acc[rt][v] + bias;
      }
    }
  } else {
#pragma unroll
    for (int rt = 0; rt < 4; ++rt) {
#pragma unroll
      for (int v = 0; v < 8; ++v) {
        const int row = base + rt * 16 + v + 8 * half;
        if (row < NN) out[(size_t)row * FD + n] = acc[rt][v] + bias;
      }
    }
  }
}

// -------------------------------------------------------------- launch ----
extern "C" void kernel_launch(void* const* d_in, const int* in_sizes, int n_in,
                              void* d_out, int out_size, void* d_ws, size_t ws_size,
                              hipStream_t stream) {
  const float* x      = (const float*)d_in[0];
  const int*   ei     = (const int*)  d_in[1];
  const int*   et     = (const int*)  d_in[2];
  const float* W_self = (const float*)d_in[3];
  const float* b_self = (const float*)d_in[4];
  const float* W1     = (const float*)d_in[5];
  const float* b1     = (const float*)d_in[6];
  const float* gamma  = (const float*)d_in[7];
  const float* beta   = (const float*)d_in[8];
  const float* W2     = (const float*)d_in[9];
  const float* b2     = (const float*)d_in[10];
  float* out = (float*)d_out;

  // Workspace layout (floats): agg/h1 buffer, then BN accumulators & params.
  const size_t buf_floats = (size_t)NR * NN * FD;          // 25,600,000
  float* buf   = (float*)d_ws;
  float* sums  = buf   + buf_floats;
  float* sumsq = sums  + NR * FD;
  float* scale = sumsq + NR * FD;
  float* shift = scale + NR * FD;

  // 1) zero agg + stat accumulators (contiguous, float4-aligned total)
  const long n4 = (long)(buf_floats + 2 * NR * FD) / 4;
  zero_kernel<<<dim3((unsigned)((n4 + 255) / 256)), dim3(256), 0, stream>>>(
      (float4*)d_ws, n4);

  // 2) edge scatter (8 edges per 256-thread block)
  scatter_kernel<<<dim3((NE + 7) / 8), dim3(256), 0, stream>>>(x, ei, et, buf);

  // 3) per-relation GEMM1 + fused BN statistics
  const int strips = (NN + ROWS - 1) / ROWS;               // 782
  const size_t lds = (size_t)LDS_FLOATS * sizeof(float);   // ~101 KB < 320 KB/WGP
  gemm1_kernel<<<dim3(strips, NR), dim3(256), lds, stream>>>(
      x, W1, b1, buf, sums, sumsq);

  // 4) finalize BN scale/shift
  bn_finalize_kernel<<<dim3(2), dim3(256), 0, stream>>>(
      sums, sumsq, gamma, beta, scale, shift);

  // 5) self-loop + per-relation GEMM2, summed into out
  gemm2_kernel<<<dim3(strips), dim3(256), lds, stream>>>(
      x, W_self, b_self, W2, b2, buf, scale, shift, out);
}